// SimpleModel_54932631716061
// MI455X (gfx1250) — compile-verified
//
#include <hip/hip_runtime.h>
#include <hip/hip_bf16.h>

#define NODES 768
#define W 32
#define MAXNB 128
#define CUT2 64.0f

typedef float v2f __attribute__((ext_vector_type(2)));
typedef float v8f __attribute__((ext_vector_type(8)));

// fast sigmoid / silu: v_exp_f32 + v_rcp_f32 (≈1 ulp), branchless
__device__ __forceinline__ float fast_sigmoid(float x) {
  return __builtin_amdgcn_rcpf(1.0f + __expf(-x));
}
__device__ __forceinline__ float fast_silu(float x) {
  return x * fast_sigmoid(x);
}

// ---------------------------------------------------------------- graph build
// One wave (32 threads) per destination node v. Deterministic ballot-ordered
// compaction of in-neighbors u (u ascending). Padded slots -> u=0, dist=0.
__global__ __launch_bounds__(32) void k_build_graph(const float* __restrict__ pos,
                                                    int* __restrict__ nbr,
                                                    int* __restrict__ cnt,
                                                    float* __restrict__ dist) {
  int v = blockIdx.x;
  int lane = threadIdx.x;
  float vx = pos[v * 3 + 0], vy = pos[v * 3 + 1], vz = pos[v * 3 + 2];
  int count = 0;
  for (int base = 0; base < NODES; base += 32) {
    int u = base + lane;
    float dx = pos[u * 3 + 0] - vx;
    float dy = pos[u * 3 + 1] - vy;
    float dz = pos[u * 3 + 2] - vz;
    float d2 = dx * dx + dy * dy + dz * dz;
    bool val = (u != v) && (d2 < CUT2);
    unsigned mask = (unsigned)__ballot(val);
    int prefix = __popc(mask & ((1u << lane) - 1u));
    int idx = count + prefix;
    if (val && idx < MAXNB) {
      nbr[v * MAXNB + idx] = u;
      dist[v * MAXNB + idx] = sqrtf(d2);
    }
    count += __popc(mask);
  }
  if (count > MAXNB) count = MAXNB;
  for (int j = count + lane; j < MAXNB; j += 32) {
    nbr[v * MAXNB + j] = 0;
    dist[v * MAXNB + j] = 0.0f;
  }
  if (lane == 0) cnt[v] = count;
}

// ---------------------------------------------------------------- x = 1
__global__ void k_init_x(float* __restrict__ x) {
  int idx = blockIdx.x * blockDim.x + threadIdx.x;
  if (idx < NODES * W) x[idx] = 1.0f;
}

// ---------------------------------------------------------------- y init
// y[v][j][c] = dist * We[c] + be[c] for real edges, 0 for padded slots.
__global__ void k_init_y(const float* __restrict__ dist, const int* __restrict__ cnt,
                         const float* __restrict__ We, const float* __restrict__ be,
                         float* __restrict__ y) {
  int idx = blockIdx.x * blockDim.x + threadIdx.x;
  if (idx >= NODES * MAXNB * W) return;
  int c = idx & (W - 1);
  int s = idx >> 5;          // slot index in [N*MAXNB)
  int v = s >> 7;            // MAXNB = 128
  int j = s & (MAXNB - 1);
  float out = 0.0f;
  if (j < cnt[v]) out = dist[s] * We[c] + be[c];
  y[idx] = out;
}

// ---------------------------------------------------------------- node gates
// esrc = x@Wsg+bsg ; edst = x@Wdg+bdg ; Bh = x@Wdu+bdu ; xs = x@Wsu+bsu
__global__ void k_node_gates(const float* __restrict__ x,
                             const float* __restrict__ Wsg, const float* __restrict__ bsg,
                             const float* __restrict__ Wdg, const float* __restrict__ bdg,
                             const float* __restrict__ Wdu, const float* __restrict__ bdu,
                             const float* __restrict__ Wsu, const float* __restrict__ bsu,
                             float* __restrict__ esrc, float* __restrict__ edst,
                             float* __restrict__ Bh, float* __restrict__ xs) {
  int idx = blockIdx.x * blockDim.x + threadIdx.x;
  if (idx >= NODES * W) return;
  int i = idx >> 5;
  int c = idx & (W - 1);
  float a0 = 0.f, a1 = 0.f, a2 = 0.f, a3 = 0.f;
#pragma unroll
  for (int k = 0; k < W; ++k) {
    float xv = x[i * W + k];
    a0 += xv * Wsg[k * W + c];
    a1 += xv * Wdg[k * W + c];
    a2 += xv * Wdu[k * W + c];
    a3 += xv * Wsu[k * W + c];
  }
  esrc[idx] = a0 + bsg[c];
  edst[idx] = a1 + bdg[c];
  Bh[idx]   = a2 + bdu[c];
  xs[idx]   = a3 + bsu[c];
}

// ---------------------------------------------------------------- edge WMMA
// For block v: m[v][j][:] = y[v][j][:] @ Weg + beg + esrc[u] + edst[v]
// sigma = valid ? sigmoid(m) : 0 ; accumulate ss[v] = sum sigma,
// ssh[v] = sum sigma * Bh[u]. Uses V_WMMA_F32_16X16X4_F32, K=32 as 8 chained
// steps, two 16-wide column tiles. 128 threads = 4 waves, each wave 2 row tiles.
__global__ __launch_bounds__(128) void k_edge_wmma(
    const float* __restrict__ y, const float* __restrict__ Weg,
    const float* __restrict__ beg, const float* __restrict__ esrc,
    const float* __restrict__ edst, const float* __restrict__ Bh,
    const int* __restrict__ nbr, const int* __restrict__ cnt,
    float* __restrict__ m, float* __restrict__ ss, float* __restrict__ ssh) {
  __shared__ int s_nbr[MAXNB];
  __shared__ float s_gate[W];
  __shared__ float s_partS[4 * W];
  __shared__ float s_partH[4 * W];

  int v = blockIdx.x;
  int tid = threadIdx.x;
  int lane = tid & 31;
  int wave = tid >> 5;
  if (tid < MAXNB) s_nbr[tid] = nbr[v * MAXNB + tid];
  if (tid < W) s_gate[tid] = edst[v * W + tid] + beg[tid];
  __syncthreads();
  int cv = cnt[v];

  int nlo = lane & 15;
  int koff = (lane >> 4) * 2;   // f32 WMMA: lanes 0-15 hold K={0,1}, lanes 16-31 K={2,3}

  // B fragments: Weg is row-major [k][n] (fi, fo)
  v2f bf_[2][8];
#pragma unroll
  for (int nt = 0; nt < 2; ++nt) {
#pragma unroll
    for (int kb = 0; kb < 8; ++kb) {
      int k0 = 4 * kb + koff;
      int n = nt * 16 + nlo;
      bf_[nt][kb].x = Weg[k0 * W + n];
      bf_[nt][kb].y = Weg[(k0 + 1) * W + n];
    }
  }

  float accS[2] = {0.f, 0.f};
  float accH[2] = {0.f, 0.f};

  for (int t = wave; t < MAXNB / 16; t += 4) {
    int rowbase = t * 16;
    const float* yr = y + ((size_t)v * MAXNB + rowbase + nlo) * W;
    v2f af[8];
#pragma unroll
    for (int kb = 0; kb < 8; ++kb)
      af[kb] = *(const v2f*)(yr + 4 * kb + koff);

    v8f c0 = {};
    v8f c1 = {};
#pragma unroll
    for (int kb = 0; kb < 8; ++kb) {
      c0 = __builtin_amdgcn_wmma_f32_16x16x4_f32(false, af[kb], false, bf_[0][kb],
                                                 (short)0, c0, false, false);
      c1 = __builtin_amdgcn_wmma_f32_16x16x4_f32(false, af[kb], false, bf_[1][kb],
                                                 (short)0, c1, false, false);
    }

    int half = lane >> 4;
#pragma unroll
    for (int nt = 0; nt < 2; ++nt) {
      int ch = nt * 16 + nlo;
      v8f cc = nt ? c1 : c0;
      float ga = s_gate[ch];
#pragma unroll
      for (int r = 0; r < 8; ++r) {
        int slot = rowbase + r + 8 * half;
        int u = s_nbr[slot];
        float mv = cc[r] + ga + esrc[u * W + ch];
        m[((size_t)v * MAXNB + slot) * W + ch] = mv;
        // branchless masked sigmoid (padded slots contribute exactly 0)
        float sg = fast_sigmoid(mv);
        sg = (slot < cv) ? sg : 0.0f;
        accS[nt] += sg;
        accH[nt] += sg * Bh[u * W + ch];
      }
    }
  }

  // combine lane halves (same channel, different row halves)
#pragma unroll
  for (int nt = 0; nt < 2; ++nt) {
    accS[nt] += __shfl_xor(accS[nt], 16, 32);
    accH[nt] += __shfl_xor(accH[nt], 16, 32);
  }
  if (lane < 16) {
#pragma unroll
    for (int nt = 0; nt < 2; ++nt) {
      s_partS[wave * W + nt * 16 + nlo] = accS[nt];
      s_partH[wave * W + nt * 16 + nlo] = accH[nt];
    }
  }
  __syncthreads();
  if (tid < W) {
    float sS = 0.f, sH = 0.f;
#pragma unroll
    for (int w = 0; w < 4; ++w) {
      sS += s_partS[w * W + tid];
      sH += s_partH[w * W + tid];
    }
    ss[v * W + tid] = sS;
    ssh[v * W + tid] = sH;
  }
}

// ---------------------------------------------------------------- h + pre-BN
__global__ void k_node_h(const float* __restrict__ ss, const float* __restrict__ ssh,
                         const float* __restrict__ xs, float* __restrict__ xt) {
  int idx = blockIdx.x * blockDim.x + threadIdx.x;
  if (idx >= NODES * W) return;
  float h = ssh[idx] / (ss[idx] + 1e-6f);
  xt[idx] = xs[idx] + h;
}

// ---------------------------------------------------------------- node BN stats
__global__ __launch_bounds__(1024) void k_node_stats(const float* __restrict__ xt,
                                                     const float* __restrict__ g,
                                                     const float* __restrict__ b,
                                                     float* __restrict__ nodeSS) {
  __shared__ float sS[1024];
  __shared__ float sQ[1024];
  int t = threadIdx.x;
  int grp = t >> 5;
  int c = t & (W - 1);
  float s = 0.f, q = 0.f;
  for (int i = grp; i < NODES; i += 32) {
    float v = xt[i * W + c];
    s += v;
    q += v * v;
  }
  sS[grp * W + c] = s;
  sQ[grp * W + c] = q;
  __syncthreads();
  for (int st = 16; st > 0; st >>= 1) {
    if (grp < st) {
      sS[grp * W + c] += sS[(grp + st) * W + c];
      sQ[grp * W + c] += sQ[(grp + st) * W + c];
    }
    __syncthreads();
  }
  if (t < W) {
    float mean = sS[t] / (float)NODES;
    float var = sQ[t] / (float)NODES - mean * mean;
    float rstd = rsqrtf(var + 1e-5f);
    float scale = g[t] * rstd;
    nodeSS[t] = scale;
    nodeSS[W + t] = b[t] - mean * scale;
  }
}

// ---------------------------------------------------------------- node finalize
__global__ void k_node_fin(float* __restrict__ x, const float* __restrict__ xt,
                           const float* __restrict__ nodeSS) {
  int idx = blockIdx.x * blockDim.x + threadIdx.x;
  if (idx >= NODES * W) return;
  int c = idx & (W - 1);
  float t = xt[idx] * nodeSS[c] + nodeSS[W + c];
  x[idx] += fast_silu(t);
}

// ---------------------------------------------------------------- edge BN stats
// One block per channel; deterministic tree reduce over all real edges.
__global__ __launch_bounds__(256) void k_edge_stats(const float* __restrict__ m,
                                                    const int* __restrict__ cnt,
                                                    const float* __restrict__ g,
                                                    const float* __restrict__ b,
                                                    float* __restrict__ eSS) {
  __shared__ float sS[256];
  __shared__ float sQ[256];
  __shared__ float sC[256];
  int c = blockIdx.x;
  int t = threadIdx.x;
  float s = 0.f, q = 0.f, n = 0.f;
  for (int sidx = t; sidx < NODES * MAXNB; sidx += 256) {
    int v = sidx >> 7;
    int j = sidx & (MAXNB - 1);
    if (j < cnt[v]) {
      float mv = m[(size_t)sidx * W + c];
      s += mv;
      q += mv * mv;
      n += 1.f;
    }
  }
  sS[t] = s; sQ[t] = q; sC[t] = n;
  __syncthreads();
  for (int st = 128; st > 0; st >>= 1) {
    if (t < st) { sS[t] += sS[t + st]; sQ[t] += sQ[t + st]; sC[t] += sC[t + st]; }
    __syncthreads();
  }
  if (t == 0) {
    float cntE = sC[0];
    float mean = sS[0] / cntE;
    float var = sQ[0] / cntE - mean * mean;
    float rstd = rsqrtf(var + 1e-5f);
    float scale = g[c] * rstd;
    eSS[c] = scale;
    eSS[W + c] = b[c] - mean * scale;
  }
}

// ---------------------------------------------------------------- edge finalize
__global__ void k_edge_fin(float* __restrict__ y, const float* __restrict__ m,
                           const int* __restrict__ cnt, const float* __restrict__ eSS) {
  int idx = blockIdx.x * blockDim.x + threadIdx.x;
  if (idx >= NODES * MAXNB * W) return;
  int c = idx & (W - 1);
  int s = idx >> 5;
  int v = s >> 7;
  int j = s & (MAXNB - 1);
  if (j < cnt[v]) {
    float t = m[idx] * eSS[c] + eSS[W + c];
    y[idx] += fast_silu(t);
  }
}

// ---------------------------------------------------------------- energy
__global__ __launch_bounds__(256) void k_energy(const float* __restrict__ x,
                                                const float* __restrict__ Wf,
                                                const float* __restrict__ bf,
                                                float* __restrict__ out) {
  __shared__ float red[256];
  int t = threadIdx.x;
  float p = 0.f;
  for (int i = t; i < NODES; i += 256) {
    float d = 0.f;
#pragma unroll
    for (int k = 0; k < W; ++k) d += x[i * W + k] * Wf[k];
    p += d;
  }
  red[t] = p;
  __syncthreads();
  for (int st = 128; st > 0; st >>= 1) {
    if (t < st) red[t] += red[t + st];
    __syncthreads();
  }
  if (t == 0) out[0] = red[0] + (float)NODES * bf[0];
}

// ================================================================ host side
struct ConvP {
  const float *Wsg, *bsg, *Wdg, *bdg, *Weg, *beg, *Wsu, *bsu, *Wdu, *bdu;
  const float *bn_n_g, *bn_n_b, *bn_e_g, *bn_e_b;
};

static ConvP conv_insertion(void* const* d, int base) {
  ConvP p;
  p.Wsg = (const float*)d[base + 0];  p.bsg = (const float*)d[base + 1];
  p.Wdg = (const float*)d[base + 2];  p.bdg = (const float*)d[base + 3];
  p.Weg = (const float*)d[base + 4];  p.beg = (const float*)d[base + 5];
  p.Wsu = (const float*)d[base + 6];  p.bsu = (const float*)d[base + 7];
  p.Wdu = (const float*)d[base + 8];  p.bdu = (const float*)d[base + 9];
  p.bn_n_g = (const float*)d[base + 10]; p.bn_n_b = (const float*)d[base + 11];
  p.bn_e_g = (const float*)d[base + 12]; p.bn_e_b = (const float*)d[base + 13];
  return p;
}

static ConvP conv_sorted(void* const* d, int base) {
  // sorted keys: Wdg,Wdu,Weg,Wsg,Wsu,bdg,bdu,beg,bn_e_b,bn_e_g,bn_n_b,bn_n_g,bsg,bsu
  ConvP p;
  p.Wdg = (const float*)d[base + 0];  p.Wdu = (const float*)d[base + 1];
  p.Weg = (const float*)d[base + 2];  p.Wsg = (const float*)d[base + 3];
  p.Wsu = (const float*)d[base + 4];  p.bdg = (const float*)d[base + 5];
  p.bdu = (const float*)d[base + 6];  p.beg = (const float*)d[base + 7];
  p.bn_e_b = (const float*)d[base + 8];  p.bn_e_g = (const float*)d[base + 9];
  p.bn_n_b = (const float*)d[base + 10]; p.bn_n_g = (const float*)d[base + 11];
  p.bsg = (const float*)d[base + 12];    p.bsu = (const float*)d[base + 13];
  return p;
}

extern "C" void kernel_launch(void* const* d_in, const int* in_sizes, int n_in,
                              void* d_out, int out_size, void* d_ws, size_t ws_size,
                              hipStream_t stream) {
  (void)n_in; (void)out_size; (void)ws_size;

  // ---- resolve input layout (dict insertion vs pytree-sorted flattening)
  const float *pos, *We, *be, *Wf, *bf;
  ConvP p1, p2;
  if (in_sizes[0] == NODES * 3) {
    pos = (const float*)d_in[0];
    if (in_sizes[3] == W * W) {
      // full insertion order: positions, We, be, conv1(14), conv2(14), Wf, bf
      We = (const float*)d_in[1]; be = (const float*)d_in[2];
      p1 = conv_insertion(d_in, 3);
      p2 = conv_insertion(d_in, 17);
      Wf = (const float*)d_in[31]; bf = (const float*)d_in[32];
    } else {
      // positions first, nested params sorted: We, Wf, be, bf, conv1(14), conv2(14)
      We = (const float*)d_in[1]; Wf = (const float*)d_in[2];
      be = (const float*)d_in[3]; bf = (const float*)d_in[4];
      p1 = conv_sorted(d_in, 5);
      p2 = conv_sorted(d_in, 19);
    }
  } else {
    // fully sorted: params (We,Wf,be,bf,conv1,conv2), then positions
    We = (const float*)d_in[0]; Wf = (const float*)d_in[1];
    be = (const float*)d_in[2]; bf = (const float*)d_in[3];
    p1 = conv_sorted(d_in, 4);
    p2 = conv_sorted(d_in, 18);
    pos = (const float*)d_in[32];
  }

  // ---- carve workspace (bump allocator, 256B aligned)
  char* wp = (char*)d_ws;
  auto alloc = [&](size_t bytes) -> void* {
    void* r = (void*)wp;
    wp += (bytes + 255) & ~(size_t)255;
    return r;
  };
  int*   nbr  = (int*)alloc((size_t)NODES * MAXNB * sizeof(int));
  int*   cnt  = (int*)alloc((size_t)NODES * sizeof(int));
  float* dist = (float*)alloc((size_t)NODES * MAXNB * sizeof(float));
  float* x    = (float*)alloc((size_t)NODES * W * sizeof(float));
  float* esrc = (float*)alloc((size_t)NODES * W * sizeof(float));
  float* edst = (float*)alloc((size_t)NODES * W * sizeof(float));
  float* Bh   = (float*)alloc((size_t)NODES * W * sizeof(float));
  float* xs   = (float*)alloc((size_t)NODES * W * sizeof(float));
  float* ssb  = (float*)alloc((size_t)NODES * W * sizeof(float));
  float* ssh  = (float*)alloc((size_t)NODES * W * sizeof(float));
  float* xt   = (float*)alloc((size_t)NODES * W * sizeof(float));
  float* nodeSS = (float*)alloc(2 * W * sizeof(float));
  float* eSS    = (float*)alloc(2 * W * sizeof(float));
  float* yb   = (float*)alloc((size_t)NODES * MAXNB * W * sizeof(float));
  float* mb   = (float*)alloc((size_t)NODES * MAXNB * W * sizeof(float));

  const int nw = NODES * W;                    // 24576
  const int blkNW = (nw + 255) / 256;          // 96
  const int nmw = NODES * MAXNB * W;           // 3145728
  const int blkNMW = (nmw + 255) / 256;        // 12288

  k_init_x<<<blkNW, 256, 0, stream>>>(x);
  k_build_graph<<<NODES, 32, 0, stream>>>(pos, nbr, cnt, dist);
  k_init_y<<<blkNMW, 256, 0, stream>>>(dist, cnt, We, be, yb);

  const ConvP* layers[2] = {&p1, &p2};
  for (int l = 0; l < 2; ++l) {
    const ConvP& p = *layers[l];
    k_node_gates<<<blkNW, 256, 0, stream>>>(x, p.Wsg, p.bsg, p.Wdg, p.bdg,
                                            p.Wdu, p.bdu, p.Wsu, p.bsu,
                                            esrc, edst, Bh, xs);
    k_edge_wmma<<<NODES, 128, 0, stream>>>(yb, p.Weg, p.beg, esrc, edst, Bh,
                                           nbr, cnt, mb, ssb, ssh);
    k_node_h<<<blkNW, 256, 0, stream>>>(ssb, ssh, xs, xt);
    k_node_stats<<<1, 1024, 0, stream>>>(xt, p.bn_n_g, p.bn_n_b, nodeSS);
    k_node_fin<<<blkNW, 256, 0, stream>>>(x, xt, nodeSS);
    k_edge_stats<<<W, 256, 0, stream>>>(mb, cnt, p.bn_e_g, p.bn_e_b, eSS);
    k_edge_fin<<<blkNMW, 256, 0, stream>>>(yb, mb, cnt, eSS);
  }

  k_energy<<<1, 256, 0, stream>>>(x, Wf, bf, (float*)d_out);
}